// GPHead_80814104641591
// MI455X (gfx1250) — compile-verified
//
#include <hip/hip_runtime.h>

typedef __attribute__((ext_vector_type(2))) float v2f;
typedef __attribute__((ext_vector_type(8))) float v8f;

namespace {
constexpr int kD  = 1024;   // feature dim
constexpr int kNs = 512;    // N_way * K_shot
constexpr int kC  = 16;     // N_way (classes)
}

// D = A(16x4 f32) * B(4x16 f32) + C(16x16 f32), wave32.
__device__ __forceinline__ v8f wmma_k4(v2f a, v2f b, v8f c) {
  return __builtin_amdgcn_wmma_f32_16x16x4_f32(
      /*neg_a=*/false, a, /*neg_b=*/false, b,
      /*c_mod=*/(short)0, c, /*reuse_a=*/false, /*reuse_b=*/false);
}

// ---------------------------------------------------------------------------
// 1) Row squared norms of Zs. One wave per row.
// ---------------------------------------------------------------------------
__global__ __launch_bounds__(256) void gp_sqnorm_kernel(const float* __restrict__ Z,
                                                        float* __restrict__ sq) {
  const int wave = threadIdx.x >> 5, lane = threadIdx.x & 31;
  const int row = blockIdx.x * 8 + wave;
  const float* p = Z + (size_t)row * kD;
  float s = 0.f;
  for (int k = lane; k < kD; k += 32) { float x = p[k]; s += x * x; }
#pragma unroll
  for (int off = 16; off > 0; off >>= 1) s += __shfl_down(s, off, 32);
  if (lane == 0) sq[row] = s;
}

// ---------------------------------------------------------------------------
// 2) Kss = exp(-max(|zi|^2+|zj|^2-2 zi.zj,0)/(2 ls^2)) + noise*I  (512x512)
//    One 16x16 tile per wave, K-loop over D via V_WMMA_F32_16X16X4_F32.
// ---------------------------------------------------------------------------
__global__ __launch_bounds__(128) void gp_kss_kernel(
    const float* __restrict__ Zs, const float* __restrict__ sqs,
    const float* __restrict__ log_ls, const float* __restrict__ log_noise,
    float* __restrict__ Kss) {
  const int tid = threadIdx.x;
  const int wave = tid >> 5, lane = tid & 31;
  const int l16 = lane & 15, half = lane >> 4;
  const int tile = blockIdx.x * 4 + wave;         // 32x32 tiles of 16x16
  const int ti = tile >> 5, tj = tile & 31;
  // A fragment: lane holds row l16, K = 2*half + {0,1}. B mirrors it.
  const float* arow = Zs + (size_t)(ti * 16 + l16) * kD + 2 * half;
  const float* brow = Zs + (size_t)(tj * 16 + l16) * kD + 2 * half;
  v8f acc = {};
  for (int k = 0; k < kD; k += 4)
    acc = wmma_k4(*(const v2f*)(arow + k), *(const v2f*)(brow + k), acc);
  const float inv2  = 0.5f / expf(2.0f * log_ls[0]);
  const float noise = expf(2.0f * log_noise[0]);
  const float sj = sqs[tj * 16 + l16];
#pragma unroll
  for (int v = 0; v < 8; ++v) {                   // C row m = v + 8*half
    const int gi = ti * 16 + v + 8 * half;
    const int gj = tj * 16 + l16;
    float d2 = fmaxf(sqs[gi] + sj - 2.0f * acc[v], 0.0f);
    float kv = expf(-d2 * inv2);
    if (gi == gj) kv += noise;
    Kss[gi * kNs + gj] = kv;
  }
}

// ---------------------------------------------------------------------------
// 3) In-place right-looking Cholesky of 512x512 SPD (lower triangle valid).
//    Single workgroup, pivot column staged in LDS.
// ---------------------------------------------------------------------------
__global__ __launch_bounds__(1024) void gp_cholesky_kernel(float* __restrict__ A) {
  const int N = kNs;
  __shared__ float s_col[kNs];
  __shared__ float s_d;
  const int tid = threadIdx.x, nthr = blockDim.x;
  const int wave = tid >> 5, lane = tid & 31, nw = nthr >> 5;
  for (int k = 0; k < N; ++k) {
    if (tid == 0) s_d = sqrtf(A[k * N + k]);
    __syncthreads();
    const float d = s_d;
    for (int i = k + 1 + tid; i < N; i += nthr) {
      float v = A[i * N + k] / d;
      A[i * N + k] = v;
      s_col[i] = v;
    }
    if (tid == 0) A[k * N + k] = d;
    __syncthreads();
    // trailing rank-1 update of the lower triangle: row per wave, cols per lane
    for (int i = k + 1 + wave; i < N; i += nw) {
      const float lik = s_col[i];
      float* Ai = A + i * N;
      for (int j = k + 1 + lane; j <= i; j += 32) Ai[j] -= lik * s_col[j];
    }
    __syncthreads();
  }
}

// ---------------------------------------------------------------------------
// 4) alpha = (L L^T)^{-1} onehot(Ys)   (16 RHS), single workgroup.
// ---------------------------------------------------------------------------
__global__ __launch_bounds__(512) void gp_solve_kernel(const float* __restrict__ L,
                                                       const int* __restrict__ Ys,
                                                       float* __restrict__ alpha) {
  const int N = kNs, C = kC;
  __shared__ float z[kNs * kC];     // 32 KB
  __shared__ float s_col[kNs];
  const int tid = threadIdx.x, nthr = blockDim.x;
  for (int idx = tid; idx < N * C; idx += nthr) {
    int i = idx / C, c = idx % C;
    z[idx] = (Ys[i] == c) ? 1.0f : 0.0f;
  }
  __syncthreads();
  // forward solve: L z = y  (right-looking; column i of L staged in LDS)
  for (int i = 0; i < N; ++i) {
    for (int r = i + tid; r < N; r += nthr) s_col[r] = L[r * N + i];
    if (tid == 0) s_col[i] = L[i * N + i];
    __syncthreads();
    if (tid < C) z[i * C + tid] /= s_col[i];
    __syncthreads();
    const int m = (N - 1 - i) * C;
    for (int idx = tid; idx < m; idx += nthr) {
      int r = i + 1 + idx / C, c = idx % C;
      z[r * C + c] -= s_col[r] * z[i * C + c];
    }
    __syncthreads();
  }
  // backward solve: L^T a = z  (uses contiguous row i of L)
  for (int i = N - 1; i >= 0; --i) {
    for (int r = tid; r <= i; r += nthr) s_col[r] = L[i * N + r];
    __syncthreads();
    if (tid < C) z[i * C + tid] /= s_col[i];
    __syncthreads();
    const int m = i * C;
    for (int idx = tid; idx < m; idx += nthr) {
      int r = idx / C, c = idx % C;
      z[r * C + c] -= s_col[r] * z[i * C + c];
    }
    __syncthreads();
  }
  for (int idx = tid; idx < N * C; idx += nthr) alpha[idx] = z[idx];
}

// ---------------------------------------------------------------------------
// 5) Fused: Kqs = rbf(Zq, Zs); f = Kqs @ alpha; softmax -> out.
//    256 threads = 8 waves; each wave owns 16 queries; support processed in
//    8 chunks of 64 (4 WMMA tiles each). K-tile spilled to padded LDS
//    (stride 65 -> conflict-free), then dense 16-class accumulation.
// ---------------------------------------------------------------------------
__global__ __launch_bounds__(256) void gp_query_kernel(
    const float* __restrict__ Zq, const float* __restrict__ Zs,
    const float* __restrict__ sqs, const float* __restrict__ alpha,
    const float* __restrict__ log_ls, float* __restrict__ out) {
  __shared__ float kS[8][16 * 65];  // per-wave 16x64 K tile, padded
  __shared__ float sqqS[8][16];     // per-wave query sq-norms

  const int tid = threadIdx.x;
  const int wave = tid >> 5, lane = tid & 31;
  const int l16 = lane & 15, half = lane >> 4;

  const float inv2 = 0.5f / expf(2.0f * log_ls[0]);
  const int qbase = blockIdx.x * 128 + wave * 16;
  const float* arow = Zq + (size_t)(qbase + l16) * kD + 2 * half;

  // query squared norms (lanes L and L+16 cover complementary K halves)
  float sp = 0.f;
  for (int k = 0; k < kD; k += 4) {
    v2f a = *(const v2f*)(arow + k);
    sp += a.x * a.x + a.y * a.y;
  }
  sp += __shfl_xor(sp, 16, 32);
  if (half == 0) sqqS[wave][l16] = sp;
  __syncthreads();

  float facc[8];
#pragma unroll
  for (int c = 0; c < 8; ++c) facc[c] = 0.f;

  for (int chunk = 0; chunk < 8; ++chunk) {
    const int sbase = chunk * 64;
    const float* brow = Zs + (size_t)(sbase + l16) * kD + 2 * half;
    v8f a0 = {}, a1 = {}, a2 = {}, a3 = {};
#pragma unroll 2
    for (int k = 0; k < kD; k += 4) {
      v2f a = *(const v2f*)(arow + k);
      a0 = wmma_k4(a, *(const v2f*)(brow + k), a0);
      a1 = wmma_k4(a, *(const v2f*)(brow + 16 * kD + k), a1);
      a2 = wmma_k4(a, *(const v2f*)(brow + 32 * kD + k), a2);
      a3 = wmma_k4(a, *(const v2f*)(brow + 48 * kD + k), a3);
    }
    const float s0 = sqs[sbase + l16];
    const float s1 = sqs[sbase + 16 + l16];
    const float s2 = sqs[sbase + 32 + l16];
    const float s3 = sqs[sbase + 48 + l16];
#pragma unroll
    for (int v = 0; v < 8; ++v) {
      const int m = v + 8 * half;              // C-tile row
      const float qn = sqqS[wave][m];
      float d0 = fmaxf(qn + s0 - 2.0f * a0[v], 0.0f);
      float d1 = fmaxf(qn + s1 - 2.0f * a1[v], 0.0f);
      float d2 = fmaxf(qn + s2 - 2.0f * a2[v], 0.0f);
      float d3 = fmaxf(qn + s3 - 2.0f * a3[v], 0.0f);
      kS[wave][m * 65 +  0 + l16] = expf(-d0 * inv2);
      kS[wave][m * 65 + 16 + l16] = expf(-d1 * inv2);
      kS[wave][m * 65 + 32 + l16] = expf(-d2 * inv2);
      kS[wave][m * 65 + 48 + l16] = expf(-d3 * inv2);
    }
    __syncthreads();
    // f[q][cb..cb+7] += K[q][s] * alpha[s][cb..cb+7]; alpha loads broadcast.
    const int cb = half * 8;
#pragma unroll 4
    for (int s = 0; s < 64; ++s) {
      const float kv = kS[wave][l16 * 65 + s];
      const float4* ap = (const float4*)(alpha + (size_t)(sbase + s) * kC + cb);
      const float4 p0 = ap[0], p1 = ap[1];
      facc[0] += kv * p0.x; facc[1] += kv * p0.y;
      facc[2] += kv * p0.z; facc[3] += kv * p0.w;
      facc[4] += kv * p1.x; facc[5] += kv * p1.y;
      facc[6] += kv * p1.z; facc[7] += kv * p1.w;
    }
    __syncthreads();
  }

  // softmax across 16 classes (8 local + 8 in lane^16)
  float mx = facc[0];
#pragma unroll
  for (int c = 1; c < 8; ++c) mx = fmaxf(mx, facc[c]);
  mx = fmaxf(mx, __shfl_xor(mx, 16, 32));
  float e[8], sum = 0.f;
#pragma unroll
  for (int c = 0; c < 8; ++c) { e[c] = expf(facc[c] - mx); sum += e[c]; }
  sum += __shfl_xor(sum, 16, 32);
  const float inv = 1.0f / sum;
  float* op = out + (size_t)(qbase + l16) * kC + half * 8;
#pragma unroll
  for (int c = 0; c < 8; ++c) op[c] = e[c] * inv;
}

// ---------------------------------------------------------------------------
extern "C" void kernel_launch(void* const* d_in, const int* in_sizes, int n_in,
                              void* d_out, int out_size, void* d_ws, size_t ws_size,
                              hipStream_t stream) {
  (void)in_sizes; (void)n_in; (void)out_size; (void)ws_size;
  const float* Zs        = (const float*)d_in[0];   // [512,1024]
  const int*   Ys        = (const int*)d_in[1];     // [512,1]
  const float* Zq        = (const float*)d_in[2];   // [65536,1024]
  const float* log_ls    = (const float*)d_in[3];   // scalar
  const float* log_noise = (const float*)d_in[4];   // scalar

  float* ws    = (float*)d_ws;
  float* Kss   = ws;                         // 512*512 floats (L after cholesky)
  float* sqs   = ws + kNs * kNs;             // 512 floats
  float* alpha = ws + kNs * kNs + kNs;       // 512*16 floats
  float* out   = (float*)d_out;              // [65536,16]

  gp_sqnorm_kernel<<<kNs / 8, 256, 0, stream>>>(Zs, sqs);
  gp_kss_kernel<<<(32 * 32) / 4, 128, 0, stream>>>(Zs, sqs, log_ls, log_noise, Kss);
  gp_cholesky_kernel<<<1, 1024, 0, stream>>>(Kss);
  gp_solve_kernel<<<1, 512, 0, stream>>>(Kss, Ys, alpha);
  gp_query_kernel<<<65536 / 128, 256, 0, stream>>>(Zq, Zs, sqs, alpha, log_ls, out);
}